// GCNLayer_75909251989599
// MI455X (gfx1250) — compile-verified
//
#include <hip/hip_runtime.h>
#include <math.h>

#define N_NODES 10000
#define N_EDGES 640000
#define D_FEAT  128
#define D_IN    512   // 4 * D_FEAT
#define D_OUT   128

typedef __attribute__((ext_vector_type(2))) float v2f;
typedef __attribute__((ext_vector_type(8))) float v8f;

// Order-preserving float atomic max via integer atomics (works for +/- values,
// init value is -inf). Avoids depending on atomicMax(float*) header support.
__device__ __forceinline__ void atomic_max_f32(float* addr, float val) {
    if (val >= 0.0f) {
        atomicMax((int*)addr, __float_as_int(val));
    } else {
        atomicMin((unsigned int*)addr, __float_as_uint(val));
    }
}

// --- K1: per-node L2 normalize * norm -> X[:,0:128]; init accumulators ------
__global__ void k_normalize_init(const float* __restrict__ h,
                                 const float* __restrict__ nrm,
                                 float* __restrict__ X,
                                 int* __restrict__ deg) {
    int node = (blockIdx.x * blockDim.x + threadIdx.x) >> 5;
    int lane = threadIdx.x & 31;
    if (node >= N_NODES) return;

    float4 v = ((const float4*)(h + (size_t)node * D_FEAT))[lane];
    float ss = v.x * v.x + v.y * v.y + v.z * v.z + v.w * v.w;
#pragma unroll
    for (int off = 16; off > 0; off >>= 1)
        ss += __shfl_xor(ss, off, 32);

    float scale = rsqrtf(fmaxf(ss, 1e-12f)) * nrm[node];

    float* Xrow = X + (size_t)node * D_IN;
    float4 hn = make_float4(v.x * scale, v.y * scale, v.z * scale, v.w * scale);
    ((float4*)(Xrow))[lane]            = hn;                                    // h_norm
    ((float4*)(Xrow + D_FEAT))[lane]   = make_float4(0.f, 0.f, 0.f, 0.f);       // agg sum
    float ni = -INFINITY;
    ((float4*)(Xrow + 2 * D_FEAT))[lane] = make_float4(ni, ni, ni, ni);         // seg max
    ((float4*)(Xrow + 3 * D_FEAT))[lane] = make_float4(0.f, 0.f, 0.f, 0.f);     // hs sum
    if (lane == 0) deg[node] = 0;
}

// --- K2: per-edge gather + scatter reductions -------------------------------
__global__ void k_edge_scatter(const int* __restrict__ src,
                               const int* __restrict__ dst,
                               float* __restrict__ X,
                               int* __restrict__ deg) {
    int e    = (blockIdx.x * blockDim.x + threadIdx.x) >> 5;
    int lane = threadIdx.x & 31;
    if (e >= N_EDGES) return;

    int s = src[e];
    int d = dst[e];

    float4 m4 = ((const float4*)(X + (size_t)s * D_IN))[lane];  // message (h_norm cols 0..127)
    float m[4] = {m4.x, m4.y, m4.z, m4.w};

    float* Xd = X + (size_t)d * D_IN + lane * 4;
#pragma unroll
    for (int j = 0; j < 4; ++j) {
        atomicAdd(Xd + D_FEAT + j, m[j]);                                    // segment sum
        atomic_max_f32(Xd + 2 * D_FEAT + j, m[j]);                           // segment max
        float hs = fminf(fmaxf(0.2f * m[j] + 0.5f, 0.0f), 1.0f);             // hard_sigmoid
        atomicAdd(Xd + 3 * D_FEAT + j, hs);                                  // hs sum
    }
    if (lane == 0) atomicAdd(deg + d, 1);
}

// --- K3: per-node finalize of the three reduced blocks ----------------------
__global__ void k_finalize(const float* __restrict__ nrm,
                           const int* __restrict__ deg,
                           float* __restrict__ X) {
    int node = (blockIdx.x * blockDim.x + threadIdx.x) >> 5;
    int lane = threadIdx.x & 31;
    if (node >= N_NODES) return;

    float n  = nrm[node];
    int   dg = deg[node];
    float inv = 1.0f / (float)(dg > 1 ? dg : 1);

    float* Xrow = X + (size_t)node * D_IN + lane * 4;
#pragma unroll
    for (int j = 0; j < 4; ++j) {
        Xrow[D_FEAT + j] *= n;                                   // agg_h = sum * norm
        float mx = Xrow[2 * D_FEAT + j];
        mx = (dg > 0) ? mx : 0.0f;                               // no in-edges -> 0
        Xrow[2 * D_FEAT + j] = mx * n;                           // max * norm
        Xrow[3 * D_FEAT + j] = Xrow[3 * D_FEAT + j] * inv * n;   // mean_hs * norm
    }
}

// --- K4: out = relu(X @ W) with V_WMMA_F32_16X16X4_F32 ----------------------
// One wave computes one 16x16 output tile. Block = 8 waves = full 128-wide N.
// A fragment (16x4 f32, 2 VGPRs): lanes 0-15 hold (K=0,K=1), lanes 16-31 hold
// (K=2,K=3) for rows M=lane&15. B fragment mirrors it for the 4x16 slab.
__global__ void k_gemm_wmma(const float* __restrict__ X,
                            const float* __restrict__ W,
                            float* __restrict__ out) {
    int m0   = blockIdx.x * 16;
    int n0   = (threadIdx.x >> 5) * 16;   // wave id -> N tile
    int lane = threadIdx.x & 31;
    int half = lane >> 4;                 // 0: K=0,1  1: K=2,3
    int l16  = lane & 15;

    const float* Arow = X + (size_t)(m0 + l16) * D_IN + 2 * half;
    const float* Bptr = W + (size_t)(2 * half) * D_OUT + n0 + l16;

    v8f c = {};
#pragma unroll 4
    for (int k = 0; k < D_IN; k += 4) {
        v2f a = *(const v2f*)(Arow + k);          // 8B aligned load (k%4==0, +2*half)
        v2f b;
        b.x = Bptr[(size_t)k * D_OUT];
        b.y = Bptr[(size_t)k * D_OUT + D_OUT];
        c = __builtin_amdgcn_wmma_f32_16x16x4_f32(
                /*neg_a=*/false, a, /*neg_b=*/false, b,
                /*c_mod=*/(short)0, c, /*reuse_a=*/false, /*reuse_b=*/false);
    }

    // C/D layout: VGPR r -> row M = r + 8*half, col N = lane&15
    float* orow = out + (size_t)(m0 + 8 * half) * D_OUT + n0 + l16;
#pragma unroll
    for (int r = 0; r < 8; ++r)
        orow[(size_t)r * D_OUT] = fmaxf(c[r], 0.0f);   // fused ReLU
}

extern "C" void kernel_launch(void* const* d_in, const int* in_sizes, int n_in,
                              void* d_out, int out_size, void* d_ws, size_t ws_size,
                              hipStream_t stream) {
    const float* h   = (const float*)d_in[0];   // [10000,128]
    const float* nrm = (const float*)d_in[1];   // [10000,1]
    const float* W   = (const float*)d_in[2];   // [512,128]
    const int*   src = (const int*)d_in[3];     // [640000]
    const int*   dst = (const int*)d_in[4];     // [640000]
    float*       out = (float*)d_out;           // [10000,128]

    float* X   = (float*)d_ws;                  // [10000,512] fused feature matrix
    int*   deg = (int*)(X + (size_t)N_NODES * D_IN);

    // K1: normalize + init accumulators (1 wave per node)
    {
        int threads = 256;
        int blocks  = (N_NODES * 32 + threads - 1) / threads;   // 1250
        k_normalize_init<<<blocks, threads, 0, stream>>>(h, nrm, X, deg);
    }
    // K2: edge gather/scatter (1 wave per edge)
    {
        int threads = 256;
        int blocks  = (N_EDGES * 32 + threads - 1) / threads;   // 80000
        k_edge_scatter<<<blocks, threads, 0, stream>>>(src, dst, X, deg);
    }
    // K3: finalize reduced blocks (1 wave per node)
    {
        int threads = 256;
        int blocks  = (N_NODES * 32 + threads - 1) / threads;   // 1250
        k_finalize<<<blocks, threads, 0, stream>>>(nrm, deg, X);
    }
    // K4: relu(X @ W) via f32 WMMA (625 M-tiles x 8 N-tile waves)
    {
        int threads = 256;                                      // 8 waves
        int blocks  = N_NODES / 16;                             // 625
        k_gemm_wmma<<<blocks, threads, 0, stream>>>(X, W, out);
    }
}